// MLoss_60782377173145
// MI455X (gfx1250) — compile-verified
//
#include <hip/hip_runtime.h>

typedef __attribute__((ext_vector_type(4))) float f4;
typedef __attribute__((ext_vector_type(2))) float v2f;
typedef __attribute__((ext_vector_type(8))) float v8f;

#define THRESH 0.5f
#define ALPHA  0.1f

// Wave32 all-lane sum. Uses V_WMMA_F32_16X16X4_F32 with B = all-ones:
// each lane's acc lands in exactly one slot of the 16x4 A tile (a1 = 0),
// so D[m][n] = sum of the accs mapped to row m, identical across columns n.
// Lane L (0-15) holds D[0..7][L], lane L (16-31) holds D[8..15][L-16];
// summing the 8 D registers then xor-shuffling by 16 gives the wave total.
__device__ __forceinline__ float wave_sum(float acc) {
#if __has_builtin(__builtin_amdgcn_wmma_f32_16x16x4_f32)
  v2f a; a.x = acc;  a.y = 0.0f;
  v2f b; b.x = 1.0f; b.y = 1.0f;
  v8f c = {0.f, 0.f, 0.f, 0.f, 0.f, 0.f, 0.f, 0.f};
  v8f d = __builtin_amdgcn_wmma_f32_16x16x4_f32(
      /*neg_a=*/false, a, /*neg_b=*/false, b,
      /*c_mod=*/(short)0, c, /*reuse_a=*/false, /*reuse_b=*/false);
  float s = ((d[0] + d[1]) + (d[2] + d[3])) + ((d[4] + d[5]) + (d[6] + d[7]));
  s += __shfl_xor(s, 16);
  return s;
#else
  for (int off = 16; off > 0; off >>= 1) acc += __shfl_xor(acc, off);
  return acc;
#endif
}

// Returns block total in thread 0 (undefined elsewhere).
__device__ __forceinline__ float block_sum(float acc) {
  __shared__ float lds[32];
  float w = wave_sum(acc);
  const int lane = threadIdx.x & 31;
  const int wid  = threadIdx.x >> 5;
  if (lane == 0) lds[wid] = w;
  __syncthreads();
  float total = 0.0f;
  if (threadIdx.x == 0) {
    const int nw = (blockDim.x + 31) >> 5;
    for (int i = 0; i < nw; ++i) total += lds[i];
  }
  return total;
}

__global__ void MLoss_main_kernel(const float* __restrict__ x,
                                  const float* __restrict__ y,
                                  float* __restrict__ partials,
                                  long long ngroups, long long ncells) {
  const long long tid    = (long long)blockIdx.x * blockDim.x + threadIdx.x;
  const long long stride = (long long)gridDim.x * blockDim.x;
  float acc = 0.0f;

  // Main path: 4 cells (20 floats / 80 bytes) per iteration, 5x b128 NT loads per tensor.
  for (long long g = tid; g < ngroups; g += stride) {
    const f4* xp = (const f4*)(x + g * 20);
    const f4* yp = (const f4*)(y + g * 20);
    f4 x0 = __builtin_nontemporal_load(xp + 0);
    f4 x1 = __builtin_nontemporal_load(xp + 1);
    f4 x2 = __builtin_nontemporal_load(xp + 2);
    f4 x3 = __builtin_nontemporal_load(xp + 3);
    f4 x4 = __builtin_nontemporal_load(xp + 4);
    f4 y0 = __builtin_nontemporal_load(yp + 0);
    f4 y1 = __builtin_nontemporal_load(yp + 1);
    f4 y2 = __builtin_nontemporal_load(yp + 2);
    f4 y3 = __builtin_nontemporal_load(yp + 3);
    f4 y4 = __builtin_nontemporal_load(yp + 4);

    f4 d0 = y0 - x0, d1 = y1 - x1, d2 = y2 - x2, d3 = y3 - x3, d4 = y4 - x4;

    // Confidence channel positions within the 20-float group: 0, 5, 10, 15.
    const float m0 = (y0.x > THRESH) ? 1.0f : 0.0f;
    const float m1 = (y1.y > THRESH) ? 1.0f : 0.0f;
    const float m2 = (y2.z > THRESH) ? 1.0f : 0.0f;
    const float m3 = (y3.w > THRESH) ? 1.0f : 0.0f;

    const float s0 = d0.x*d0.x + d0.y*d0.y + d0.z*d0.z + d0.w*d0.w + d1.x*d1.x;
    const float s1 = d1.y*d1.y + d1.z*d1.z + d1.w*d1.w + d2.x*d2.x + d2.y*d2.y;
    const float s2 = d2.z*d2.z + d2.w*d2.w + d3.x*d3.x + d3.y*d3.y + d3.z*d3.z;
    const float s3 = d3.w*d3.w + d4.x*d4.x + d4.y*d4.y + d4.z*d4.z + d4.w*d4.w;

    const float c0 = x0.x, c1 = x1.y, c2 = x2.z, c3 = x3.w;

    acc += m0 * s0 + ALPHA * (1.0f - m0) * c0 * c0;
    acc += m1 * s1 + ALPHA * (1.0f - m1) * c1 * c1;
    acc += m2 * s2 + ALPHA * (1.0f - m2) * c2 * c2;
    acc += m3 * s3 + ALPHA * (1.0f - m3) * c3 * c3;
  }

  // Generic tail (ncells % 4 != 0) — empty for this problem size.
  const long long rem = ncells - ngroups * 4;
  if (tid < rem) {
    const long long cell = ngroups * 4 + tid;
    const float* xc = x + cell * 5;
    const float* yc = y + cell * 5;
    const float m = (yc[0] > THRESH) ? 1.0f : 0.0f;
    float s = 0.0f;
#pragma unroll
    for (int cch = 0; cch < 5; ++cch) {
      const float dd = yc[cch] - xc[cch];
      s += dd * dd;
    }
    acc += m * s + ALPHA * (1.0f - m) * xc[0] * xc[0];
  }

  const float bsum = block_sum(acc);   // all threads reach here (EXEC all-ones at WMMA)
  if (threadIdx.x == 0) partials[blockIdx.x] = bsum;
}

__global__ void MLoss_final_kernel(const float* __restrict__ partials, int n,
                                   float* __restrict__ out) {
  float acc = 0.0f;
  for (int i = threadIdx.x; i < n; i += blockDim.x) acc += partials[i];
  const float total = block_sum(acc);
  if (threadIdx.x == 0) out[0] = total;
}

extern "C" void kernel_launch(void* const* d_in, const int* in_sizes, int n_in,
                              void* d_out, int out_size, void* d_ws, size_t ws_size,
                              hipStream_t stream) {
  const float* x = (const float*)d_in[0];
  const float* y = (const float*)d_in[1];
  float* out = (float*)d_out;
  float* partials = (float*)d_ws;

  const long long n       = (long long)in_sizes[0];
  const long long ncells  = n / 5;
  const long long ngroups = ncells / 4;

  const int threads = 256;
  int blocks = 1024;
  // d_ws must hold one float per block.
  const long long maxBlocks = (long long)(ws_size / sizeof(float));
  if (maxBlocks > 0 && blocks > maxBlocks) blocks = (int)maxBlocks;
  if (blocks < 1) blocks = 1;

  MLoss_main_kernel<<<blocks, threads, 0, stream>>>(x, y, partials, ngroups, ncells);
  MLoss_final_kernel<<<1, 256, 0, stream>>>(partials, blocks, out);
}